// RocketFeaturesPytorch_83133386982115
// MI455X (gfx1250) — compile-verified
//
#include <hip/hip_runtime.h>
#include <hip/hip_bf16.h>

#define NKER   10000
#define SEQL   512
#define NB     8
#define KSM    11
#define OSTRIDE (2 * NKER)

// ---- CDNA5 async global->LDS path (guarded so it compiles on any toolchain) ----
#if defined(__HIP_DEVICE_COMPILE__) && __has_builtin(__builtin_amdgcn_global_load_async_to_lds_b128)
#define USE_ASYNC_COPY 1
#else
#define USE_ASYNC_COPY 0
#endif

#if USE_ASYNC_COPY
typedef int v4i_t __attribute__((vector_size(16)));
typedef __attribute__((address_space(1))) v4i_t* gv4i_t;  // global (AS1) v4i*
typedef __attribute__((address_space(3))) v4i_t* lv4i_t;  // LDS    (AS3) v4i*
#endif

__device__ __forceinline__ void wait_async_then_barrier() {
#if USE_ASYNC_COPY
#if __has_builtin(__builtin_amdgcn_s_wait_asynccnt)
    __builtin_amdgcn_s_wait_asynccnt(0);
#else
    asm volatile("s_wait_asynccnt 0" ::: "memory");
#endif
#endif
    __syncthreads();
}

// Kernel 1: PAD_MAX = max(base).  (pad==0 occurs, so max(base) == PAD_MAX.)
__global__ void pad_max_kernel(const int* __restrict__ base, int* __restrict__ ws) {
    __shared__ int red[256];
    int m = 0;
    for (int i = threadIdx.x; i < NKER; i += 256) m = max(m, base[i]);
    red[threadIdx.x] = m;
    __syncthreads();
    for (int s = 128; s > 0; s >>= 1) {
        if ((int)threadIdx.x < s) red[threadIdx.x] = max(red[threadIdx.x], red[threadIdx.x + s]);
        __syncthreads();
    }
    if (threadIdx.x == 0) ws[0] = red[0];
}

// Kernel 2: one wave32 per rocket kernel; block stages x (8x512 f32) in LDS.
__global__ __launch_bounds__(256) void rocket_kernel(
    const float* __restrict__ x,     // (8,1,512)
    const float* __restrict__ W,     // (10000,1,11), zero-padded taps
    const float* __restrict__ bias,  // (10000,)
    const int*   __restrict__ base,  // (10000,)
    const int*   __restrict__ dil,   // (10000,)
    const int*   __restrict__ lo,    // (10000,)
    const int*   __restrict__ ws,    // ws[0] = PAD_MAX
    float*       __restrict__ out)   // (8, 20000)
{
    __shared__ float xs[NB * SEQL];  // 16 KB

    // ---- stage x (8x512 f32) into LDS via CDNA5 async DMA ----
#if USE_ASYNC_COPY
    {
        const int tid = threadIdx.x;
#pragma unroll
        for (int c = 0; c < 4; ++c) {
            const int q = tid + c * 256;  // float4 chunk id, 1024 chunks total
            __builtin_amdgcn_global_load_async_to_lds_b128(
                (gv4i_t)(x + q * 4),
                (lv4i_t)(xs + q * 4),
                0, 0);
        }
    }
#else
    for (int i = threadIdx.x; i < NB * SEQL; i += 256) xs[i] = x[i];
#endif
    wait_async_then_barrier();

    const int pad_max = ws[0];
    int k = (int)blockIdx.x * 8 + ((int)threadIdx.x >> 5);
    k = __builtin_amdgcn_readfirstlane(k);   // wave-uniform -> scalar loads below
    if (k >= NKER) return;

    const int lane = (int)threadIdx.x & 31;
    const int   kb = base[k];
    const int   kd = dil[k];
    const int   kl = lo[k];
    const float kbias = bias[k];

    float w[KSM];
#pragma unroll
    for (int j = 0; j < KSM; ++j) w[j] = W[k * KSM + j];

    float mx[NB];
    int   ct[NB];
#pragma unroll
    for (int b = 0; b < NB; ++b) { mx[b] = -__builtin_inff(); ct[b] = 0; }

    const int ub0 = kb - pad_max;  // unpadded index base: idx = ub0 + t + j*dil

    for (int t0 = 0; t0 < kl; t0 += 32) {
        const int  t  = t0 + lane;
        const bool tv = (t < kl);

        // Index math + padding mask once per t-tile; mask folded into the
        // (per-lane) weight so the 8-batch inner loop is pure ds_load + fma.
        int   idx[KSM];
        float wm[KSM];
#pragma unroll
        for (int j = 0; j < KSM; ++j) {
            const int raw = ub0 + t + j * kd;
            wm[j]  = (((unsigned)raw) < (unsigned)SEQL) ? w[j] : 0.0f;  // padding -> 0
            idx[j] = min(max(raw, 0), SEQL - 1);  // clamp: no EXEC divergence on ds_load
        }

#pragma unroll
        for (int b = 0; b < NB; ++b) {
            float acc = kbias;
#pragma unroll
            for (int j = 0; j < KSM; ++j) {
                acc = fmaf(wm[j], xs[b * SEQL + idx[j]], acc);
            }
            if (tv) {
                mx[b] = fmaxf(mx[b], acc);
                ct[b] += (acc > 0.0f) ? 1 : 0;
            }
        }
    }

    // wave32 butterfly reduction per batch, lane 0 writes [max, ppv]
#pragma unroll
    for (int b = 0; b < NB; ++b) {
        float m = mx[b];
        int   c = ct[b];
#pragma unroll
        for (int off = 16; off > 0; off >>= 1) {
            m = fmaxf(m, __shfl_xor(m, off, 32));
            c += __shfl_xor(c, off, 32);
        }
        if (lane == 0) {
            out[b * OSTRIDE + 2 * k]     = m;
            out[b * OSTRIDE + 2 * k + 1] = (float)c / (float)kl;
        }
    }
}

extern "C" void kernel_launch(void* const* d_in, const int* in_sizes, int n_in,
                              void* d_out, int out_size, void* d_ws, size_t ws_size,
                              hipStream_t stream) {
    const float* x    = (const float*)d_in[0];   // (8,1,512)
    const float* W    = (const float*)d_in[1];   // (10000,1,11)
    const float* bias = (const float*)d_in[2];   // (10000,)
    const int*   base = (const int*)d_in[3];     // (10000,)
    const int*   dil  = (const int*)d_in[4];     // (10000,)
    const int*   lo   = (const int*)d_in[5];     // (10000,)
    int*   ws  = (int*)d_ws;
    float* out = (float*)d_out;

    pad_max_kernel<<<1, 256, 0, stream>>>(base, ws);
    rocket_kernel<<<(NKER + 7) / 8, 256, 0, stream>>>(x, W, bias, base, dil, lo, ws, out);
}